// LSTM_46050639347908
// MI455X (gfx1250) — compile-verified
//
#include <hip/hip_runtime.h>
#include <math.h>

#define T_DIM 32768
#define I_DIM 128
#define H_DIM 512
#define G_DIM 2048      // 4*H

// persistent LSTM config
#define NWG   32        // workgroups in recurrence (each owns 16 hidden units = 64 gate rows)
#define NTHR  64        // threads per WG (2 waves, wave32)
#define ROWS  64        // gate rows per WG
#define WSTR  516       // padded LDS row stride in floats (516%64==4 -> mild conflicts, 16B aligned)

typedef float v2f __attribute__((ext_vector_type(2)));
typedef float v8f __attribute__((ext_vector_type(8)));

// ---------------- Kernel 1: v[k] = sum_m Wout[m]*W1[m,k] ; v[512] = sum_m Wout[m]*b1[m] ----
__global__ void prep_v_kernel(const float* __restrict__ W1,
                              const float* __restrict__ b1,
                              const float* __restrict__ Wout,
                              float* __restrict__ v)
{
  int tid = threadIdx.x;
  for (int k = tid; k < H_DIM; k += 64) {
    float s = 0.f;
    for (int m = 0; m < 25; ++m) s += Wout[m] * W1[m * H_DIM + k];
    v[k] = s;
  }
  if (tid == 0) {
    float s = 0.f;
    for (int m = 0; m < 25; ++m) s += Wout[m] * b1[m];
    v[H_DIM] = s;
  }
}

// ---------------- Kernel 2: xg[t][g] = x[t,:]·W_ih[g,:] + b_ih[g]+b_hh[g] ------------------
// f32 WMMA 16x16x4; each wave computes a 16(M=t) x 64(N=g) strip over K=128.
// A (16x4 f32):  lanes 0-15 hold M=lane, K={0,1}; lanes 16-31 hold M=lane-16, K={2,3}
// B (4x16 f32):  lanes 0-15 hold N=lane, K={0,1}; lanes 16-31 hold N=lane-16, K={2,3}
// D (16x16 f32): VGPR r -> M = (lane>=16 ? 8 : 0)+r, N = lane&15
__global__ void xg_gemm_kernel(const float* __restrict__ x,
                               const float* __restrict__ Wih,
                               const float* __restrict__ bih,
                               const float* __restrict__ bhh,
                               float* __restrict__ xg)
{
  const int wave = threadIdx.x >> 5;
  const int lane = threadIdx.x & 31;
  const int t0 = (blockIdx.x * 8 + wave) * 16;
  const int g0 = blockIdx.y * 64;
  const int l16 = lane & 15;
  const int hi  = lane >> 4;       // 0 or 1
  const int kb  = hi * 2;          // K offset inside the 4-slice

  const float* xrow = x + (size_t)(t0 + l16) * I_DIM + kb;
  const float* wp[4];
  wp[0] = Wih + (size_t)(g0 +  0 + l16) * I_DIM + kb;
  wp[1] = wp[0] + 16 * I_DIM;
  wp[2] = wp[0] + 32 * I_DIM;
  wp[3] = wp[0] + 48 * I_DIM;

  v8f acc[4] = {v8f{}, v8f{}, v8f{}, v8f{}};

  #pragma unroll 4
  for (int k0 = 0; k0 < I_DIM; k0 += 4) {
    v2f av = *(const v2f*)(xrow + k0);
    #pragma unroll
    for (int nt = 0; nt < 4; ++nt) {
      v2f bv = *(const v2f*)(wp[nt] + k0);
      acc[nt] = __builtin_amdgcn_wmma_f32_16x16x4_f32(
          false, av, false, bv, (short)0, acc[nt], false, false);
    }
  }

  #pragma unroll
  for (int nt = 0; nt < 4; ++nt) {
    int g = g0 + nt * 16 + l16;
    float bias = bih[g] + bhh[g];
    #pragma unroll
    for (int r = 0; r < 8; ++r) {
      int t = t0 + hi * 8 + r;
      __builtin_nontemporal_store(acc[nt][r] + bias, &xg[(size_t)t * G_DIM + g]);
    }
  }
}

// ---------------- Kernel 3: persistent LSTM recurrence + fused output projection ----------
__device__ __forceinline__ float sigmoidf_(float z) { return 1.f / (1.f + __expf(-z)); }

__global__ void __launch_bounds__(NTHR, 1)
lstm_kernel(const float* __restrict__ xg,
            const float* __restrict__ Whh,
            const float* __restrict__ vbuf,   // v[0..511], c0 at [512]
            float* __restrict__ hbuf,         // double buffer: 2*512 f32 (zeroed by memset)
            unsigned* __restrict__ syncv,     // [0]=arrive counter, [32]=generation (zeroed)
            float* __restrict__ out)
{
  extern __shared__ float smem[];
  float* W_s = smem;                 // ROWS*WSTR  (~129 KB of the 320 KB WGP LDS)
  float* h_s = W_s + ROWS * WSTR;    // 512
  float* v_s = h_s + H_DIM;          // 512
  float* g_s = v_s + H_DIM;          // 64 gate pre-activations
  float* r_s = g_s + ROWS;           // 2 reduction slots

  const int tid = threadIdx.x;
  const int wg  = blockIdx.x;
  const int j0  = wg * 16;                   // hidden slice [j0, j0+16)
  const int gi  = tid >> 4;                  // gate index 0..3 (i,f,g,o)
  const int jj  = tid & 15;
  const int g_row = gi * H_DIM + j0 + jj;    // my gate row in [0,2048)

  // Stage my W_hh slice into LDS (padded rows: conflict-avoiding b128 reads later)
  for (int idx = tid; idx < ROWS * H_DIM; idx += NTHR) {
    int r = idx >> 9;
    int k = idx & (H_DIM - 1);
    int gr = (r >> 4) * H_DIM + j0 + (r & 15);
    W_s[r * WSTR + k] = Whh[(size_t)gr * H_DIM + k];
  }
  for (int k = tid; k < H_DIM; k += NTHR) v_s[k] = vbuf[k];
  const float c0v = vbuf[H_DIM];
  __syncthreads();

  float c = 0.f;
  unsigned* cntp = syncv;
  unsigned* genp = syncv + 32;   // separate cacheline

  for (int t = 0; t < T_DIM; ++t) {
    const int rp = (t + 1) & 1;  // buffer holding h_{t-1}
    const int wpar = t & 1;      // buffer to receive h_t

    // independent of h: fetch my xg element early, prefetch next step
    float xgv = __builtin_nontemporal_load(&xg[(size_t)t * G_DIM + g_row]);
    if (t + 1 < T_DIM)
      __builtin_prefetch(&xg[(size_t)(t + 1) * G_DIM + g_row], 0, 0);

    // broadcast h_{t-1}: device-scope loads (bypass stale L0), into LDS
    #pragma unroll
    for (int i = 0; i < 8; ++i) {
      int k = i * NTHR + tid;
      h_s[k] = __hip_atomic_load(&hbuf[rp * H_DIM + k],
                                 __ATOMIC_RELAXED, __HIP_MEMORY_SCOPE_AGENT);
    }
    __syncthreads();

    // WG0 overlaps the output projection of h_{t-1}: out[t-1] = h·v + c0
    if (wg == 0 && t > 0) {
      float p = 0.f;
      #pragma unroll
      for (int i = 0; i < 8; ++i) { int k = i * NTHR + tid; p += h_s[k] * v_s[k]; }
      #pragma unroll
      for (int off = 16; off > 0; off >>= 1) p += __shfl_down(p, off, 32);
      if ((tid & 31) == 0) r_s[tid >> 5] = p;
      __syncthreads();
      if (tid == 0) out[t - 1] = r_s[0] + r_s[1] + c0v;
    }

    // GEMV: one gate row per thread, 512 MACs from LDS (b128 W + broadcast h)
    float acc = xgv;
    const float* wr = W_s + tid * WSTR;
    #pragma unroll 8
    for (int k = 0; k < H_DIM; k += 4) {
      float4 w4 = *(const float4*)(wr + k);
      float4 h4 = *(const float4*)(h_s + k);
      acc += w4.x * h4.x + w4.y * h4.y + w4.z * h4.z + w4.w * h4.w;
    }
    g_s[tid] = acc;
    __syncthreads();

    // elementwise cell update for my 16 hidden units
    if (tid < 16) {
      float ig = sigmoidf_(g_s[ 0 + tid]);
      float fg = sigmoidf_(g_s[16 + tid]);
      float gg = tanhf    (g_s[32 + tid]);
      float og = sigmoidf_(g_s[48 + tid]);
      c = fg * c + ig * gg;
      float h = og * tanhf(c);
      __hip_atomic_store(&hbuf[wpar * H_DIM + j0 + tid], h,
                         __ATOMIC_RELAXED, __HIP_MEMORY_SCOPE_AGENT);
    }

    // device-wide barrier (generation counter, s_sleep spin)
    __threadfence();
    __syncthreads();
    if (tid == 0) {
      unsigned g = __hip_atomic_load(genp, __ATOMIC_RELAXED, __HIP_MEMORY_SCOPE_AGENT);
      unsigned a = __hip_atomic_fetch_add(cntp, 1u, __ATOMIC_ACQ_REL, __HIP_MEMORY_SCOPE_AGENT);
      if (a == NWG - 1) {
        __hip_atomic_store(cntp, 0u, __ATOMIC_RELAXED, __HIP_MEMORY_SCOPE_AGENT);
        __hip_atomic_store(genp, g + 1u, __ATOMIC_RELEASE, __HIP_MEMORY_SCOPE_AGENT);
      } else {
        while (__hip_atomic_load(genp, __ATOMIC_ACQUIRE, __HIP_MEMORY_SCOPE_AGENT) == g)
          __builtin_amdgcn_s_sleep(2);
      }
    }
    __syncthreads();
  }

  // last output: projection of h_{T-1}
  if (wg == 0) {
    const int rp = (T_DIM - 1) & 1;
    #pragma unroll
    for (int i = 0; i < 8; ++i) {
      int k = i * NTHR + tid;
      h_s[k] = __hip_atomic_load(&hbuf[rp * H_DIM + k],
                                 __ATOMIC_RELAXED, __HIP_MEMORY_SCOPE_AGENT);
    }
    __syncthreads();
    float p = 0.f;
    #pragma unroll
    for (int i = 0; i < 8; ++i) { int k = i * NTHR + tid; p += h_s[k] * v_s[k]; }
    #pragma unroll
    for (int off = 16; off > 0; off >>= 1) p += __shfl_down(p, off, 32);
    if ((tid & 31) == 0) r_s[tid >> 5] = p;
    __syncthreads();
    if (tid == 0) out[T_DIM - 1] = r_s[0] + r_s[1] + c0v;
  }
}

// ---------------------------------- host launcher ----------------------------------------
extern "C" void kernel_launch(void* const* d_in, const int* in_sizes, int n_in,
                              void* d_out, int out_size, void* d_ws, size_t ws_size,
                              hipStream_t stream)
{
  const float* x    = (const float*)d_in[0];
  const float* Wih  = (const float*)d_in[1];
  const float* Whh  = (const float*)d_in[2];
  const float* bih  = (const float*)d_in[3];
  const float* bhh  = (const float*)d_in[4];
  const float* W1   = (const float*)d_in[5];
  const float* b1   = (const float*)d_in[6];
  const float* Wout = (const float*)d_in[7];
  float* out = (float*)d_out;

  const size_t XG_BYTES = (size_t)T_DIM * G_DIM * sizeof(float);   // 256 MB
  char* ws = (char*)d_ws;
  float*    xg    = (float*)ws;
  float*    hbuf  = (float*)(ws + XG_BYTES);           // 2*512 f32
  float*    vbuf  = (float*)(ws + XG_BYTES + 4096);    // 513 f32
  unsigned* syncv = (unsigned*)(ws + XG_BYTES + 8192); // counter + gen

  // zero h double-buffer, v region, and sync counters (deterministic per launch)
  hipMemsetAsync(ws + XG_BYTES, 0, 8192 + 256, stream);

  prep_v_kernel<<<1, 64, 0, stream>>>(W1, b1, Wout, vbuf);
  xg_gemm_kernel<<<dim3(T_DIM / 128, G_DIM / 64), 256, 0, stream>>>(x, Wih, bih, bhh, xg);

  const int lds_bytes = (int)((ROWS * WSTR + 2 * H_DIM + ROWS + 2) * sizeof(float));
  hipFuncSetAttribute(reinterpret_cast<const void*>(lstm_kernel),
                      hipFuncAttributeMaxDynamicSharedMemorySize, lds_bytes);
  lstm_kernel<<<NWG, NTHR, lds_bytes, stream>>>(xg, Whh, vbuf, hbuf, syncv, out);

  (void)in_sizes; (void)n_in; (void)out_size; (void)ws_size;
}